// HistogramLoss_53764400611560
// MI455X (gfx1250) — compile-verified
//
#include <hip/hip_runtime.h>
#include <hip/hip_bf16.h>

typedef __attribute__((ext_vector_type(2))) float v2f;
typedef __attribute__((ext_vector_type(8))) float v8f;

#define N_PTS   1024
#define DIMS    128
#define NBINS   151
#define TILE    16
#define NT      (N_PTS / TILE)          // 64 tile rows/cols
#define NTILES  (NT * (NT + 1) / 2)     // 2080 upper-triangular tiles

// ---------------------------------------------------------------------------
// workspace layout (floats):
//   [0 .. 150]    raw hist_pos
//   [151 .. 301]  raw hist_neg
//   [302 .. 303]  pos_count, neg_count (as unsigned bits)
// ---------------------------------------------------------------------------

__global__ void hist_zero_kernel(float* ws) {
    int i = blockIdx.x * blockDim.x + threadIdx.x;
    if (i < 2 * NBINS + 2) ws[i] = 0.0f;
}

__global__ __launch_bounds__(32)
void hist_tile_kernel(const float* __restrict__ feats,
                      const int*   __restrict__ classes,
                      float* __restrict__ histPos,
                      float* __restrict__ histNeg,
                      unsigned int* __restrict__ counts) {
    __shared__ float hP[NBINS];
    __shared__ float hN[NBINS];
    __shared__ unsigned int cP, cN;

    const int lane = threadIdx.x;        // 0..31 (wave32)
    const int lo   = lane & 15;
    const int hi   = lane >> 4;

    for (int i = lane; i < NBINS; i += 32) { hP[i] = 0.0f; hN[i] = 0.0f; }
    if (lane == 0) { cP = 0u; cN = 0u; }
    __syncthreads();

    const float step  = 2.0f / (float)(NBINS - 1);
    const float rstep = (float)(NBINS - 1) / 2.0f;

    for (int t = blockIdx.x; t < NTILES; t += gridDim.x) {
        // decode linear upper-triangular tile index -> (rt, ct), ct >= rt
        int rt = 0, rem = t;
        while (rem >= NT - rt) { rem -= NT - rt; ++rt; }
        const int ct = rt + rem;

        const int rowBase = rt * TILE;
        const int colBase = ct * TILE;

        // ---- 16x16 fp32 Gram tile via V_WMMA_F32_16X16X4_F32, K swept 0..127
        // A frag: lane holds A[lo][kk + 2*hi + v]  (v = 0,1)
        // B frag: lane holds B[kk + 2*hi + v][lo] = feats[colBase+lo][...]
        v8f c = {};
        const float* aRow = feats + (rowBase + lo) * DIMS + 2 * hi;
        const float* bRow = feats + (colBase + lo) * DIMS + 2 * hi;
        #pragma unroll
        for (int kk = 0; kk < DIMS; kk += 4) {
            v2f a = *(const v2f*)(aRow + kk);
            v2f b = *(const v2f*)(bRow + kk);
            c = __builtin_amdgcn_wmma_f32_16x16x4_f32(
                    false, a, false, b, (short)0, c, false, false);
        }

        // ---- bin the 8 elements this lane owns:
        //      element r -> (m, n) = (r + 8*hi, lo)
        #pragma unroll
        for (int r = 0; r < 8; ++r) {
            const int gi = rowBase + r + 8 * hi;
            const int gj = colBase + lo;
            if (gi < gj) {                         // strict upper triangle
                const float s  = c[r];
                const bool  eq = (classes[gi] == classes[gj]);

                // reference: delta = floor((s+1)/step)*step - 1
                const float u     = (s + 1.0f) * rstep;
                const float fl    = floorf(u);
                const int   jb    = (int)fl;
                const float delta = fl * step - 1.0f;
                const float wa    = (s - delta) * rstep;   // weight for bin jb+1
                const float wb    = 1.0f - wa;             // weight for bin jb

                float* h = eq ? hP : hN;
                if (jb >= 0 && jb < NBINS)         atomicAdd(&h[jb],     wb);
                if (jb + 1 >= 0 && jb + 1 < NBINS) atomicAdd(&h[jb + 1], wa);
                if (eq) atomicAdd(&cP, 1u); else atomicAdd(&cN, 1u);
            }
        }
    }
    __syncthreads();

    // flush per-block LDS histogram to global workspace
    for (int i = lane; i < NBINS; i += 32) {
        if (hP[i] != 0.0f) atomicAdd(&histPos[i], hP[i]);
        if (hN[i] != 0.0f) atomicAdd(&histNeg[i], hN[i]);
    }
    if (lane == 0) {
        if (cP) atomicAdd(&counts[0], cP);
        if (cN) atomicAdd(&counts[1], cN);
    }
}

__global__ void hist_finalize_kernel(const float* __restrict__ histPos,
                                     const float* __restrict__ histNeg,
                                     const unsigned int* __restrict__ counts,
                                     float* __restrict__ out) {
    if (threadIdx.x == 0 && blockIdx.x == 0) {
        const float ps = (float)counts[0];
        const float ns = (float)counts[1];
        float cdf = 0.0f, loss = 0.0f;
        for (int j = 0; j < NBINS; ++j) {
            cdf  += histPos[j] / ps;
            loss += (histNeg[j] / ns) * cdf;
        }
        out[0] = loss;
    }
}

extern "C" void kernel_launch(void* const* d_in, const int* in_sizes, int n_in,
                              void* d_out, int out_size, void* d_ws, size_t ws_size,
                              hipStream_t stream) {
    const float* feats   = (const float*)d_in[0];   // [1024,128] f32
    const int*   classes = (const int*)d_in[1];     // [1024] i32
    float*       out     = (float*)d_out;           // scalar f32

    float*        ws      = (float*)d_ws;
    float*        histPos = ws;                     // 151
    float*        histNeg = ws + NBINS;             // 151
    unsigned int* counts  = (unsigned int*)(ws + 2 * NBINS);  // 2

    hist_zero_kernel<<<(2 * NBINS + 2 + 255) / 256, 256, 0, stream>>>(ws);
    hist_tile_kernel<<<260, 32, 0, stream>>>(feats, classes, histPos, histNeg, counts);
    hist_finalize_kernel<<<1, 32, 0, stream>>>(histPos, histNeg, counts, out);
}